// WaveAttention_40776419508928
// MI455X (gfx1250) — compile-verified
//
#include <hip/hip_runtime.h>
#include <hip/hip_bf16.h>

typedef __attribute__((ext_vector_type(2))) float v2f;
typedef __attribute__((ext_vector_type(8))) float v8f;

__device__ __forceinline__ v8f wmma4(v2f a, v2f b, v8f c) {
  // D = A(16x4 f32) * B(4x16 f32) + C(16x16 f32)
  return __builtin_amdgcn_wmma_f32_16x16x4_f32(
      /*neg_a=*/false, a, /*neg_b=*/false, b,
      /*c_mod=*/(short)0, c, /*reuse_a=*/false, /*reuse_b=*/false);
}

__device__ __forceinline__ v8f v8zero() {
  v8f z = {0.f, 0.f, 0.f, 0.f, 0.f, 0.f, 0.f, 0.f};
  return z;
}

__device__ __forceinline__ v2f mk2(float x, float y) {
  v2f r; r.x = x; r.y = y; return r;
}

// ---------------------------------------------------------------------------
// Generic fp32 WMMA GEMM: C[M,N] = A[M,K](lda) * W[N,K]^T + bias, C stride ldc
// grid = (M/64, N/64), block = 128 (4 waves). Wave w computes rows w*16..+16,
// all 64 columns of the block tile. A/B tiles staged in LDS k-pair-interleaved
// (v2f) so every WMMA fragment is a single aligned ds_load_b64.
// ---------------------------------------------------------------------------
__global__ __launch_bounds__(128) void gemm_f32_wmma(
    const float* __restrict__ A, int lda,
    const float* __restrict__ W,   // [N][K] row-major
    const float* __restrict__ bias,
    float* __restrict__ C, int ldc, int K) {
  __shared__ v2f As2[64][9];   // [row][k-pair 0..7] + pad
  __shared__ v2f Bs2[8][65];   // [k-pair][n 0..63] + pad

  const int m0 = blockIdx.x * 64;
  const int n0 = blockIdx.y * 64;
  const int t = threadIdx.x;
  const int w = t >> 5;
  const int lane = t & 31;
  const int col = lane & 15;
  const int half = lane >> 4;

  v8f acc[4];
#pragma unroll
  for (int i = 0; i < 4; ++i) acc[i] = v8zero();

  const int rowL = t >> 1;          // 0..63
  const int kkL = (t & 1) * 8;      // 0 or 8
  const int kpL = kkL >> 1;         // 0 or 4 (pair index)
  const float* aptr = A + (size_t)(m0 + rowL) * lda + kkL;
  const float* wptr = W + (size_t)(n0 + rowL) * K + kkL;

  for (int k0 = 0; k0 < K; k0 += 16) {
    __syncthreads();
    // stage A 64x16 (pair-interleaved)
    float4 a0 = *(const float4*)(aptr + k0);
    float4 a1 = *(const float4*)(aptr + k0 + 4);
    As2[rowL][kpL + 0] = mk2(a0.x, a0.y);
    As2[rowL][kpL + 1] = mk2(a0.z, a0.w);
    As2[rowL][kpL + 2] = mk2(a1.x, a1.y);
    As2[rowL][kpL + 3] = mk2(a1.z, a1.w);
    // stage B 16x64 transposed (pair-interleaved along k)
    float4 b0 = *(const float4*)(wptr + k0);
    float4 b1 = *(const float4*)(wptr + k0 + 4);
    Bs2[kpL + 0][rowL] = mk2(b0.x, b0.y);
    Bs2[kpL + 1][rowL] = mk2(b0.z, b0.w);
    Bs2[kpL + 2][rowL] = mk2(b1.x, b1.y);
    Bs2[kpL + 3][rowL] = mk2(b1.z, b1.w);
    if (k0 + 16 < K) {  // prefetch next tiles (global_prefetch_b8)
      __builtin_prefetch(aptr + k0 + 16, 0, 0);
      __builtin_prefetch(wptr + k0 + 16, 0, 0);
    }
    __syncthreads();

#pragma unroll
    for (int kb2 = 0; kb2 < 8; kb2 += 2) {   // 4 k-steps of 4
      const v2f a = As2[w * 16 + col][kb2 + half];
#pragma unroll
      for (int nt = 0; nt < 4; ++nt) {
        const v2f b = Bs2[kb2 + half][nt * 16 + col];
        acc[nt] = wmma4(a, b, acc[nt]);
      }
    }
  }

#pragma unroll
  for (int nt = 0; nt < 4; ++nt) {
    const int n = n0 + nt * 16 + col;
    const float bv = bias[n];
#pragma unroll
    for (int r = 0; r < 8; ++r) {
      const int m = m0 + w * 16 + half * 8 + r;
      C[(size_t)m * ldc + n] = acc[nt][r] + bv;
    }
  }
}

// ---------------------------------------------------------------------------
// Wfilt (O=512,C=512,3,3) -> Wt2[tap][c-pair][o] as v2f {c even, c odd}
// ---------------------------------------------------------------------------
__global__ void wfilt_transpose(const float* __restrict__ Wf, v2f* __restrict__ Wt2) {
  const unsigned idx = blockIdx.x * 256u + threadIdx.x;
  if (idx >= 9u * 256u * 512u) return;
  const unsigned o = idx & 511u;
  const unsigned c2 = (idx >> 9) & 255u;
  const unsigned tap = idx >> 17;
  Wt2[idx] = mk2(Wf[((size_t)o * 512 + 2 * c2) * 9 + tap],
                 Wf[((size_t)o * 512 + 2 * c2 + 1) * 9 + tap]);
}

// ---------------------------------------------------------------------------
// Haar DWT: xr[bg][p=8x8][128] -> dwt[bg][s=4x4][band*128+c]
// ---------------------------------------------------------------------------
__global__ __launch_bounds__(128) void dwt_kernel(const float* __restrict__ xr,
                                                  float* __restrict__ dwt) {
  const int bg = blockIdx.x;
  const int c = threadIdx.x; // 128 channels
  const float* base = xr + (size_t)bg * 64 * 128;
#pragma unroll 1
  for (int s = 0; s < 16; ++s) {
    const int u = s >> 2, v = s & 3;
    const float a = base[((2 * u) * 8 + 2 * v) * 128 + c];
    const float b = base[((2 * u) * 8 + 2 * v + 1) * 128 + c];
    const float cc = base[((2 * u + 1) * 8 + 2 * v) * 128 + c];
    const float d = base[((2 * u + 1) * 8 + 2 * v + 1) * 128 + c];
    float* o = dwt + ((size_t)bg * 16 + s) * 512;
    o[c]       = 0.5f * (a + b + cc + d);
    o[128 + c] = 0.5f * (a + b - cc - d);
    o[256 + c] = 0.5f * (a - b + cc - d);
    o[384 + c] = 0.5f * (a - b - cc + d);
  }
}

// ---------------------------------------------------------------------------
// 3x3 conv (pad 1) on (bg, 512ch, 4x4) via WMMA. Input dwt[bg][s][512],
// weights Wt2[tap][c-pair][o] (v2f), output convo[bg][s][512].
// grid = (2048/4, 512/64), block = 128 (4 waves). Wave w handles o-tile w,
// 4 images (M tiles) per block. Zero LDS row implements padding taps.
// ---------------------------------------------------------------------------
__global__ __launch_bounds__(128) void conv3x3_wmma(
    const float* __restrict__ dwt, const v2f* __restrict__ Wt2,
    const float* __restrict__ bfilt, float* __restrict__ convo) {
  __shared__ float Xs[4][17][130]; // [img][spatial(16)+zero-row][128ch chunk]
  const int bg0 = blockIdx.x * 4;
  const int o0 = blockIdx.y * 64;
  const int t = threadIdx.x;
  const int w = t >> 5;
  const int lane = t & 31;
  const int col = lane & 15;   // A: M row (out spatial); B/C: N col (out ch)
  const int half = lane >> 4;

  v8f acc[4];
#pragma unroll
  for (int i = 0; i < 4; ++i) acc[i] = v8zero();

  // zero-padding row (index 16) per image
  {
    const int img = t >> 5;
    const int c4 = (t & 31) * 4;
#pragma unroll
    for (int i = 0; i < 4; ++i) Xs[img][16][c4 + i] = 0.f;
  }

  const int u = col >> 2, v = col & 3;

  for (int c0 = 0; c0 < 512; c0 += 128) {
    __syncthreads();
    {
      const int img = t >> 5;
      const int s = (t >> 1) & 15;
      const int coff = (t & 1) * 64;
      const float* src = dwt + ((size_t)(bg0 + img) * 16 + s) * 512 + c0 + coff;
      float* dst = &Xs[img][s][coff];
#pragma unroll
      for (int i = 0; i < 64; i += 4) {
        float4 q = *(const float4*)(src + i);
        dst[i] = q.x; dst[i + 1] = q.y; dst[i + 2] = q.z; dst[i + 3] = q.w;
      }
    }
    __syncthreads();

#pragma unroll 1
    for (int tap = 0; tap < 9; ++tap) {
      const int di = tap / 3 - 1, dj = tap % 3 - 1;
      const int su = u + di, sv = v + dj;
      const int spos = (su >= 0 && su < 4 && sv >= 0 && sv < 4) ? (su * 4 + sv) : 16;
      const v2f* wbase2 = Wt2 + ((size_t)tap * 256 + (c0 >> 1)) * 512 + o0 + w * 16 + col;
#pragma unroll 4
      for (int kb2 = 0; kb2 < 64; kb2 += 2) {   // 128 channels = 64 pairs
        const v2f b = wbase2[(size_t)(kb2 + half) * 512];
#pragma unroll
        for (int img = 0; img < 4; ++img) {
          const v2f a = *(const v2f*)&Xs[img][spos][(kb2 + half) * 2];
          acc[img] = wmma4(a, b, acc[img]);
        }
      }
    }
  }

  const float bv = bfilt[o0 + w * 16 + col];
#pragma unroll
  for (int img = 0; img < 4; ++img) {
#pragma unroll
    for (int r = 0; r < 8; ++r) {
      const int s = half * 8 + r;
      convo[((size_t)(bg0 + img) * 16 + s) * 512 + o0 + w * 16 + col] = acc[img][r] + bv;
    }
  }
}

// ---------------------------------------------------------------------------
// Haar IDWT: convo[bg][s][512] -> cat[row=bg*64+p][640] columns 512..639
// ---------------------------------------------------------------------------
__global__ __launch_bounds__(128) void idwt_kernel(const float* __restrict__ convo,
                                                   float* __restrict__ cat) {
  const int bg = blockIdx.x;
  const int c = threadIdx.x; // 128
#pragma unroll 1
  for (int s = 0; s < 16; ++s) {
    const float* in = convo + ((size_t)bg * 16 + s) * 512;
    const float ll = in[c], lh = in[128 + c], hl = in[256 + c], hh = in[384 + c];
    const float y00 = 0.5f * (ll + lh + hl + hh);
    const float y01 = 0.5f * (ll + lh - hl - hh);
    const float y10 = 0.5f * (ll - lh + hl - hh);
    const float y11 = 0.5f * (ll - lh - hl + hh);
    const int u = s >> 2, v = s & 3;
    const size_t rb = (size_t)bg * 64;
    cat[(rb + (2 * u) * 8 + 2 * v) * 640 + 512 + c] = y00;
    cat[(rb + (2 * u) * 8 + 2 * v + 1) * 640 + 512 + c] = y01;
    cat[(rb + (2 * u + 1) * 8 + 2 * v) * 640 + 512 + c] = y10;
    cat[(rb + (2 * u + 1) * 8 + 2 * v + 1) * 640 + 512 + c] = y11;
  }
}

// ---------------------------------------------------------------------------
// LayerNorm over last dim (512): convo[bg][s][512] -> xdn[bg][s][512]
// ---------------------------------------------------------------------------
__global__ __launch_bounds__(128) void ln_kernel(const float* __restrict__ xin,
                                                 const float* __restrict__ gamma,
                                                 const float* __restrict__ beta,
                                                 float* __restrict__ xout) {
  const int bg = blockIdx.x;
  const int t = threadIdx.x;
  const int w = t >> 5;
  const int lane = t & 31;
#pragma unroll 1
  for (int s = w * 4; s < w * 4 + 4; ++s) {
    const float* row = xin + ((size_t)bg * 16 + s) * 512;
    float vbuf[16];
    float s1 = 0.f, s2 = 0.f;
#pragma unroll
    for (int i = 0; i < 16; ++i) {
      const float vv = row[lane + 32 * i];
      vbuf[i] = vv;
      s1 += vv;
      s2 += vv * vv;
    }
#pragma unroll
    for (int off = 16; off >= 1; off >>= 1) {
      s1 += __shfl_xor(s1, off, 32);
      s2 += __shfl_xor(s2, off, 32);
    }
    const float mean = s1 * (1.0f / 512.0f);
    const float var = s2 * (1.0f / 512.0f) - mean * mean;
    const float rs = rsqrtf(var + 1e-5f);
    float* out = xout + ((size_t)bg * 16 + s) * 512;
#pragma unroll
    for (int i = 0; i < 16; ++i) {
      const int c = lane + 32 * i;
      out[c] = (vbuf[i] - mean) * rs * gamma[c] + beta[c];
    }
  }
}

// ---------------------------------------------------------------------------
// Attention per (bg): 16 heads, P=64 queries, SP=16 keys, DH=32.
// S^T = (k*scale) @ q^T so softmax over keys is lane-local + one shuffle.
// O^T = v^T @ attn^T, then out2^T = O^T + O^T @ PB^T (rel-pos bias).
// attn probs stored pair-interleaved (scp, v2f) so O-GEMM B-frags are single
// ds_load_b64; sc pad = 66 keeps rel-pos A-frags 8B aligned v2f loads.
// ---------------------------------------------------------------------------
__global__ __launch_bounds__(128) void attn_wmma(
    const float* __restrict__ qb,   // [bg*64][512]
    const float* __restrict__ kvb,  // [bg*16][1024] (k | v)
    const float* __restrict__ bias_table, // [225][16]
    const int* __restrict__ rel_index,    // [64][64]
    float* __restrict__ cat) {
  __shared__ float sc[4][32][66];  // per-wave scratch (O^T / out2^T)
  __shared__ v2f scp[4][8][66];    // per-wave attn^T, j-pair-interleaved
  const int bg = blockIdx.x;
  const int t = threadIdx.x;
  const int w = t >> 5;
  const int lane = t & 31;
  const int col = lane & 15;
  const int half = lane >> 4;
  const float scale = 0.044194173824159216f; // 512^-0.5

#pragma unroll 1
  for (int h = w * 4; h < w * 4 + 4; ++h) {
    // ---- S^T (16 keys x 64 queries) = (k*scale) @ q^T ----
    v8f st[4];
#pragma unroll
    for (int i = 0; i < 4; ++i) st[i] = v8zero();
#pragma unroll
    for (int kb = 0; kb < 32; kb += 4) {
      const int kk = kb + half * 2;
      v2f a = *(const v2f*)(kvb + ((size_t)bg * 16 + col) * 1024 + h * 32 + kk);
      a.x *= scale; a.y *= scale;
#pragma unroll
      for (int nt = 0; nt < 4; ++nt) {
        const int iq = nt * 16 + col;
        const v2f b = *(const v2f*)(qb + ((size_t)bg * 64 + iq) * 512 + h * 32 + kk);
        st[nt] = wmma4(a, b, st[nt]);
      }
    }
    // ---- softmax over j (16 keys): lane holds 8 j's, partner lane^16 rest
#pragma unroll
    for (int nt = 0; nt < 4; ++nt) {
      float m = st[nt][0];
#pragma unroll
      for (int r = 1; r < 8; ++r) m = fmaxf(m, st[nt][r]);
      m = fmaxf(m, __shfl_xor(m, 16, 32));
      float p[8], sum = 0.f;
#pragma unroll
      for (int r = 0; r < 8; ++r) { p[r] = __expf(st[nt][r] - m); sum += p[r]; }
      sum += __shfl_xor(sum, 16, 32);
      const float inv = 1.0f / sum;
#pragma unroll
      for (int r2 = 0; r2 < 4; ++r2)  // j-pairs are lane-local
        scp[w][half * 4 + r2][nt * 16 + col] = mk2(p[2 * r2] * inv, p[2 * r2 + 1] * inv);
    }
    // ---- O^T (32 x 64) = v^T @ attn^T ----
    v8f ot[2][4];
#pragma unroll
    for (int mt = 0; mt < 2; ++mt)
#pragma unroll
      for (int nt = 0; nt < 4; ++nt) ot[mt][nt] = v8zero();
#pragma unroll
    for (int kb2 = 0; kb2 < 8; kb2 += 2) {   // 16 keys = 8 pairs
      const int kj = kb2 * 2 + half * 2;
      v2f a[2];
#pragma unroll
      for (int mt = 0; mt < 2; ++mt) {
        const int d = mt * 16 + col;
        a[mt].x = kvb[((size_t)bg * 16 + kj) * 1024 + 512 + h * 32 + d];
        a[mt].y = kvb[((size_t)bg * 16 + kj + 1) * 1024 + 512 + h * 32 + d];
      }
#pragma unroll
      for (int nt = 0; nt < 4; ++nt) {
        const v2f b = scp[w][kb2 + half][nt * 16 + col];
#pragma unroll
        for (int mt = 0; mt < 2; ++mt) ot[mt][nt] = wmma4(a[mt], b, ot[mt][nt]);
      }
    }
    // stash O^T for A-fragment reads
#pragma unroll
    for (int mt = 0; mt < 2; ++mt)
#pragma unroll
      for (int nt = 0; nt < 4; ++nt)
#pragma unroll
        for (int r = 0; r < 8; ++r)
          sc[w][mt * 16 + half * 8 + r][nt * 16 + col] = ot[mt][nt][r];
    // ---- out2^T = O^T + O^T @ PB^T  (K=64 over queries j) ----
#pragma unroll 1
    for (int kb = 0; kb < 64; kb += 4) {
      const int kj = kb + half * 2;
      v2f a[2];
#pragma unroll
      for (int mt = 0; mt < 2; ++mt)
        a[mt] = *(const v2f*)&sc[w][mt * 16 + col][kj];
#pragma unroll
      for (int nt = 0; nt < 4; ++nt) {
        const int iq = nt * 16 + col;
        const v2f b = mk2(bias_table[rel_index[iq * 64 + kj] * 16 + h],
                          bias_table[rel_index[iq * 64 + kj + 1] * 16 + h]);
#pragma unroll
        for (int mt = 0; mt < 2; ++mt) ot[mt][nt] = wmma4(a[mt], b, ot[mt][nt]);
      }
    }
    // stash + coalesced writeout: lane = d (0..31)
#pragma unroll
    for (int mt = 0; mt < 2; ++mt)
#pragma unroll
      for (int nt = 0; nt < 4; ++nt)
#pragma unroll
        for (int r = 0; r < 8; ++r)
          sc[w][mt * 16 + half * 8 + r][nt * 16 + col] = ot[mt][nt][r];
#pragma unroll 1
    for (int iq = 0; iq < 64; ++iq)
      cat[((size_t)bg * 64 + iq) * 640 + h * 32 + lane] = sc[w][lane][iq];
  }
}

// ---------------------------------------------------------------------------
extern "C" void kernel_launch(void* const* d_in, const int* in_sizes, int n_in,
                              void* d_out, int out_size, void* d_ws, size_t ws_size,
                              hipStream_t stream) {
  const float* x     = (const float*)d_in[0];
  const float* Wq    = (const float*)d_in[1];
  const float* bq    = (const float*)d_in[2];
  const float* Wred  = (const float*)d_in[3];
  const float* bred  = (const float*)d_in[4];
  const float* Wfilt = (const float*)d_in[5];
  const float* bfilt = (const float*)d_in[6];
  const float* gamma = (const float*)d_in[7];
  const float* beta  = (const float*)d_in[8];
  const float* Wkv   = (const float*)d_in[9];
  const float* bkv   = (const float*)d_in[10];
  const float* Wproj = (const float*)d_in[11];
  const float* bproj = (const float*)d_in[12];
  const float* Wmerge= (const float*)d_in[13];
  const float* bmerge= (const float*)d_in[14];
  const float* btab  = (const float*)d_in[15];
  const int*   ridx  = (const int*)d_in[16];
  float* out = (float*)d_out;
  float* ws = (float*)d_ws;

  const size_t ROWS = 8ull * 256 * 64;   // 131072
  size_t off = 0;
  float* Wt      = ws + off; off += 9ull * 512 * 512;     // 2.36M (stored as v2f pairs)
  float* q_buf   = ws + off; off += ROWS * 512;           // 67.1M
  float* xr_buf  = ws + off; off += ROWS * 128;           // 16.8M
  float* dwt_buf = ws + off; off += 2048ull * 16 * 512;   // 16.8M
  float* conv_o  = ws + off; off += 2048ull * 16 * 512;   // 16.8M
  float* xdn     = ws + off; off += 2048ull * 16 * 512;   // 16.8M
  float* kv_buf  = ws + off; off += 2048ull * 16 * 1024;  // 33.6M
  float* cat_buf = ws + off; off += ROWS * 640;           // 83.9M
  float* proj_b  = ws + off; off += ROWS * 512;           // 67.1M

  // q = x @ Wq^T + bq
  gemm_f32_wmma<<<dim3(2048, 8), 128, 0, stream>>>(x, 512, Wq, bq, q_buf, 512, 512);
  // xr = x @ Wred^T + bred
  gemm_f32_wmma<<<dim3(2048, 2), 128, 0, stream>>>(x, 512, Wred, bred, xr_buf, 128, 512);
  // Haar DWT
  dwt_kernel<<<2048, 128, 0, stream>>>(xr_buf, dwt_buf);
  // weight transpose for conv (pair-interleaved)
  wfilt_transpose<<<(9 * 256 * 512) / 256, 256, 0, stream>>>(Wfilt, (v2f*)Wt);
  // 3x3 conv
  conv3x3_wmma<<<dim3(512, 8), 128, 0, stream>>>(dwt_buf, (const v2f*)Wt, bfilt, conv_o);
  // IDWT -> cat columns 512..639
  idwt_kernel<<<2048, 128, 0, stream>>>(conv_o, cat_buf);
  // LayerNorm
  ln_kernel<<<2048, 128, 0, stream>>>(conv_o, gamma, beta, xdn);
  // kv = xdn @ Wkv^T + bkv
  gemm_f32_wmma<<<dim3(512, 16), 128, 0, stream>>>(xdn, 512, Wkv, bkv, kv_buf, 1024, 512);
  // attention + rel-pos -> cat columns 0..511
  attn_wmma<<<2048, 128, 0, stream>>>(q_buf, kv_buf, btab, ridx, cat_buf);
  // proj = cat @ Wproj^T + bproj (K = 640)
  gemm_f32_wmma<<<dim3(2048, 8), 128, 0, stream>>>(cat_buf, 640, Wproj, bproj, proj_b, 512, 640);
  // out = proj @ Wmerge^T + bmerge
  gemm_f32_wmma<<<dim3(2048, 8), 128, 0, stream>>>(proj_b, 512, Wmerge, bmerge, out, 512, 512);
  (void)in_sizes; (void)n_in; (void)out_size; (void)ws_size;
}